// MambaLayer_90263032693008
// MI455X (gfx1250) — compile-verified
//
#include <hip/hip_runtime.h>
#include <hip/hip_bf16.h>
#include <math.h>

#define L_SEQ   1024
#define DMODEL  768
#define DINNER  1536
#define DSTATE  64
#define DCONV   4
#define DTRANK  48
#define N_XZ    (2 * DINNER)            // 3072
#define N_DBL   (DTRANK + 2 * DSTATE)   // 176

typedef float v2f __attribute__((ext_vector_type(2)));
typedef float v4f __attribute__((ext_vector_type(4)));
typedef float v8f __attribute__((ext_vector_type(8)));
typedef int   v4i __attribute__((ext_vector_type(4)));

#define AS1 __attribute__((address_space(1)))
#define AS3 __attribute__((address_space(3)))

#if __has_builtin(__builtin_amdgcn_global_load_async_to_lds_b128)
#define HAVE_ASYNC_LDS 1
#else
#define HAVE_ASYNC_LDS 0
#endif

__device__ __forceinline__ float silu_f(float v) { return v / (1.f + expf(-v)); }

__device__ __forceinline__ void async_wait_all() {
#if HAVE_ASYNC_LDS
#if __has_builtin(__builtin_amdgcn_s_wait_asynccnt)
  __builtin_amdgcn_s_wait_asynccnt(0);
#else
  asm volatile("s_wait_asynccnt 0x0" ::: "memory");
#endif
#endif
}

// ---------------------------------------------------------------------------
// LayerNorm: one block per row of x (1024 rows, 768 cols)
// ---------------------------------------------------------------------------
__global__ void __launch_bounds__(256) ln_kernel(const float* __restrict__ x,
                                                 const float* __restrict__ g,
                                                 const float* __restrict__ b,
                                                 float* __restrict__ xn) {
  const int l = blockIdx.x;
  const float* row = x + l * DMODEL;
  float s = 0.f, s2 = 0.f;
  for (int i = threadIdx.x; i < DMODEL; i += 256) {
    float v = row[i];
    s += v;
    s2 += v * v;
  }
  for (int o = 16; o > 0; o >>= 1) {  // wave32 reduction
    s  += __shfl_xor(s,  o, 32);
    s2 += __shfl_xor(s2, o, 32);
  }
  __shared__ float ps[8], ps2[8];
  const int w = threadIdx.x >> 5;
  if ((threadIdx.x & 31) == 0) { ps[w] = s; ps2[w] = s2; }
  __syncthreads();
  s = 0.f; s2 = 0.f;
  #pragma unroll
  for (int i = 0; i < 8; ++i) { s += ps[i]; s2 += ps2[i]; }
  const float mu  = s * (1.f / DMODEL);
  const float var = s2 * (1.f / DMODEL) - mu * mu;
  const float rs  = rsqrtf(var + 1e-5f);
  float* orow = xn + l * DMODEL;
  for (int i = threadIdx.x; i < DMODEL; i += 256)
    orow[i] = (row[i] - mu) * rs * g[i] + b[i];
}

// ---------------------------------------------------------------------------
// Blocked fp32 WMMA GEMM. Block = 256 threads (8 waves) -> 128x64 C tile.
// B panel (32 x 64 per chunk) staged into LDS with CDNA5 async global->LDS
// copies (ASYNCcnt), double-buffered so the next chunk's fill overlaps the
// current chunk's WMMAs. Each wave register-blocks 4 N-tiles so one A
// fragment feeds 4 v_wmma_f32_16x16x4_f32 ops. All strides compile-time.
//
// LDS row stride = 80 floats: lanes 0-15 (row k) and 16-31 (row k+2) then
// read disjoint bank sets (n and n+32 mod 64) -> no bank conflicts.
// ---------------------------------------------------------------------------
#define BROW 80
#define KC   32

template<int LDB>
__device__ __forceinline__ void stage_b_chunk(const float* __restrict__ B, int k0, int tn,
                                              float* buf, int tid) {
  #pragma unroll
  for (int p = 0; p < 2; ++p) {
    const int r = p * 16 + (tid >> 4);
    const int c = (tid & 15) << 2;
    const float* src = B + (k0 + r) * LDB + tn + c;
    float* dst = buf + r * BROW + c;
#if HAVE_ASYNC_LDS
    __builtin_amdgcn_global_load_async_to_lds_b128((AS1 v4i*)src, (AS3 v4i*)dst,
                                                   /*offset=*/0, /*cpol=*/0);
#else
    *(v4f*)dst = *(const v4f*)src;
#endif
  }
}

template<int LDA, int LDB, int LDC, int K>
__global__ void __launch_bounds__(256) wmma_gemm_blk(const float* __restrict__ A,
                                                     const float* __restrict__ B,
                                                     float* __restrict__ C,
                                                     int tilesN /* N/64 */) {
  __shared__ float bbuf[2][KC * BROW];

  const int tid  = threadIdx.x;
  const int wave = tid >> 5;
  const int lane = tid & 31;
  const int half = lane >> 4;   // selects K pair {0,1} vs {2,3}
  const int mr   = lane & 15;

  const int tn = (blockIdx.x % tilesN) * 64;
  const int tm = (blockIdx.x / tilesN) * 128;

  const float* Arow = A + (tm + wave * 16 + mr) * LDA + 2 * half;

  v8f acc[4] = {};

  stage_b_chunk<LDB>(B, 0, tn, bbuf[0], tid);
  async_wait_all();
  __syncthreads();

  constexpr int NCH = K / KC;
  for (int ch = 0; ch < NCH; ++ch) {
    const int cur = ch & 1;
    if (ch + 1 < NCH) stage_b_chunk<LDB>(B, (ch + 1) * KC, tn, bbuf[cur ^ 1], tid);

    const float* ap = Arow + ch * KC;
    const float* bp = bbuf[cur] + (2 * half) * BROW + mr;
    __builtin_prefetch(ap + KC, 0, 0);  // next A chunk toward L2/WGP$

    #pragma unroll
    for (int k0 = 0; k0 < KC; k0 += 4) {
      v2f a;
      a.x = ap[k0];
      a.y = ap[k0 + 1];
      const float* br = bp + k0 * BROW;
      #pragma unroll
      for (int t = 0; t < 4; ++t) {
        v2f b;
        b.x = br[t * 16];
        b.y = br[t * 16 + BROW];
        acc[t] = __builtin_amdgcn_wmma_f32_16x16x4_f32(
            /*neg_a=*/false, a, /*neg_b=*/false, b,
            /*c_mod=*/(short)0, acc[t], /*reuse_a=*/false, /*reuse_b=*/false);
      }
    }
    async_wait_all();   // own async fills of the other buffer are done
    __syncthreads();    // make them visible to all waves / free buffer we read
  }

  float* Cw = C + (tm + wave * 16 + half * 8) * LDC + tn + mr;
  #pragma unroll
  for (int r = 0; r < 8; ++r) {
    #pragma unroll
    for (int t = 0; t < 4; ++t) Cw[r * LDC + t * 16] = acc[t][r];
  }
}

// ---------------------------------------------------------------------------
// Simple wave-per-16x16-tile WMMA GEMM for the small N=176 projection.
// Compile-time strides -> immediate address offsets in the unrolled loop.
// ---------------------------------------------------------------------------
template<int LDA, int LDB, int LDC, int K>
__global__ void __launch_bounds__(256) wmma_gemm_tile(const float* __restrict__ A,
                                                      const float* __restrict__ B,
                                                      float* __restrict__ C,
                                                      int tilesN, int totalTiles) {
  const int tile = (int)((blockIdx.x * blockDim.x + threadIdx.x) >> 5);
  if (tile >= totalTiles) return;  // wave-uniform

  const int tm   = (tile / tilesN) << 4;
  const int tn   = (tile % tilesN) << 4;
  const int lane = threadIdx.x & 31;
  const int half = lane >> 4;
  const int mr   = lane & 15;

  const float* Arow = A + (tm + mr) * LDA + 2 * half;
  const float* Bcol = B + (2 * half) * LDB + tn + mr;

  v8f acc = {};
  #pragma unroll 8
  for (int k0 = 0; k0 < K; k0 += 4) {
    v2f a, b;
    a.x = Arow[k0];
    a.y = Arow[k0 + 1];
    b.x = Bcol[k0 * LDB];
    b.y = Bcol[k0 * LDB + LDB];
    acc = __builtin_amdgcn_wmma_f32_16x16x4_f32(false, a, false, b, (short)0, acc,
                                                false, false);
  }

  float* Cout = C + (tm + (half << 3)) * LDC + tn + mr;
  #pragma unroll
  for (int r = 0; r < 8; ++r) Cout[r * LDC] = acc[r];
}

// ---------------------------------------------------------------------------
// Depthwise causal conv (k=4) + bias + SiLU. xi = xz[:, :DINNER].
// ---------------------------------------------------------------------------
__global__ void __launch_bounds__(256) conv_silu_kernel(const float* __restrict__ xz,
                                                        const float* __restrict__ cw,
                                                        const float* __restrict__ cb,
                                                        float* __restrict__ xc) {
  const int idx = blockIdx.x * 256 + threadIdx.x;
  if (idx >= L_SEQ * DINNER) return;
  const int d = idx % DINNER;
  const int l = idx / DINNER;
  float acc = cb[d];
  #pragma unroll
  for (int k = 0; k < DCONV; ++k) {
    const int ls = l + k - (DCONV - 1);
    if (ls >= 0) acc = fmaf(xz[ls * N_XZ + d], cw[d * DCONV + k], acc);
  }
  xc[idx] = silu_f(acc);
}

// ---------------------------------------------------------------------------
// dt = softplus(dt_raw @ W_dt + b_dt); K=48 -> VALU fmaf chain.
// ---------------------------------------------------------------------------
__global__ void __launch_bounds__(256) dt_softplus_kernel(const float* __restrict__ dbl,
                                                          const float* __restrict__ W_dt,
                                                          const float* __restrict__ b_dt,
                                                          float* __restrict__ dt) {
  const int idx = blockIdx.x * 256 + threadIdx.x;
  if (idx >= L_SEQ * DINNER) return;
  const int d = idx % DINNER;
  const int l = idx / DINNER;
  const float* row = dbl + l * N_DBL;
  float acc = b_dt[d];
  #pragma unroll 8
  for (int r = 0; r < DTRANK; ++r) acc = fmaf(row[r], W_dt[r * DINNER + d], acc);
  dt[idx] = (acc > 20.f) ? acc : log1pf(expf(acc));
}

// ---------------------------------------------------------------------------
// Selective scan. One wave32 per d-channel; lane holds h[d,lane], h[d,lane+32].
// ---------------------------------------------------------------------------
__global__ void __launch_bounds__(256) scan_kernel(const float* __restrict__ dt,
                                                   const float* __restrict__ dbl,
                                                   const float* __restrict__ xc,
                                                   const float* __restrict__ A_log,
                                                   float* __restrict__ ys) {
  const int wave = (int)((blockIdx.x * 256 + threadIdx.x) >> 5);
  const int lane = threadIdx.x & 31;
  if (wave >= DINNER) return;
  const int d = wave;

  const float A0 = -expf(A_log[d * DSTATE + lane]);
  const float A1 = -expf(A_log[d * DSTATE + lane + 32]);
  float h0 = 0.f, h1 = 0.f;

  for (int l = 0; l < L_SEQ; ++l) {
    const float dtv = dt[l * DINNER + d];
    const float xv  = xc[l * DINNER + d];
    const float* bc = dbl + l * N_DBL + DTRANK;   // B cols 48..111, C cols 112..175
    const float B0 = bc[lane];
    const float B1 = bc[lane + 32];
    const float C0 = bc[DSTATE + lane];
    const float C1 = bc[DSTATE + lane + 32];
    const float u  = dtv * xv;
    h0 = fmaf(expf(dtv * A0), h0, B0 * u);
    h1 = fmaf(expf(dtv * A1), h1, B1 * u);
    float y = fmaf(h0, C0, h1 * C1);
    #pragma unroll
    for (int o = 16; o > 0; o >>= 1) y += __shfl_xor(y, o, 32);
    if (lane == 0) ys[l * DINNER + d] = y;
  }
}

// ---------------------------------------------------------------------------
// y = (ys + xc * D_skip) * silu(z), in-place on ys. z = xz[:, DINNER:].
// ---------------------------------------------------------------------------
__global__ void __launch_bounds__(256) gate_kernel(const float* __restrict__ xz,
                                                   const float* __restrict__ xc,
                                                   const float* __restrict__ D_skip,
                                                   float* __restrict__ y) {
  const int idx = blockIdx.x * 256 + threadIdx.x;
  if (idx >= L_SEQ * DINNER) return;
  const int d = idx % DINNER;
  const int l = idx / DINNER;
  const float z = xz[l * N_XZ + DINNER + d];
  y[idx] = fmaf(xc[idx], D_skip[d], y[idx]) * silu_f(z);
}

// ---------------------------------------------------------------------------
extern "C" void kernel_launch(void* const* d_in, const int* in_sizes, int n_in,
                              void* d_out, int out_size, void* d_ws, size_t ws_size,
                              hipStream_t stream) {
  const float* x      = (const float*)d_in[0];
  const float* ln_g   = (const float*)d_in[1];
  const float* ln_b   = (const float*)d_in[2];
  const float* W_in   = (const float*)d_in[3];
  const float* conv_w = (const float*)d_in[4];
  const float* conv_b = (const float*)d_in[5];
  const float* W_x    = (const float*)d_in[6];
  const float* W_dt   = (const float*)d_in[7];
  const float* b_dt   = (const float*)d_in[8];
  const float* A_log  = (const float*)d_in[9];
  const float* D_skip = (const float*)d_in[10];
  const float* W_out  = (const float*)d_in[11];
  float* out = (float*)d_out;

  float* ws  = (float*)d_ws;
  float* xn  = ws;                       // 1024*768
  float* xz  = xn  + L_SEQ * DMODEL;     // 1024*3072
  float* xc  = xz  + L_SEQ * N_XZ;       // 1024*1536
  float* dbl = xc  + L_SEQ * DINNER;     // 1024*176
  float* dt  = dbl + L_SEQ * N_DBL;      // 1024*1536
  float* yb  = dt  + L_SEQ * DINNER;     // 1024*1536

  const int elemBlocks = (L_SEQ * DINNER) / 256;  // 6144

  // 1. LayerNorm
  ln_kernel<<<L_SEQ, 256, 0, stream>>>(x, ln_g, ln_b, xn);

  // 2. xz = xn @ W_in   (1024x768 @ 768x3072): (1024/128)*(3072/64) = 384 blocks
  wmma_gemm_blk<DMODEL, N_XZ, N_XZ, DMODEL>
      <<<8 * 48, 256, 0, stream>>>(xn, W_in, xz, 48);

  // 3. depthwise conv + SiLU
  conv_silu_kernel<<<elemBlocks, 256, 0, stream>>>(xz, conv_w, conv_b, xc);

  // 4. dbl = xc @ W_x   (1024x1536 @ 1536x176): 64*11 wave tiles
  wmma_gemm_tile<DINNER, N_DBL, N_DBL, DINNER>
      <<<(64 * 11) / 8, 256, 0, stream>>>(xc, W_x, dbl, 11, 64 * 11);

  // 5. dt = softplus(dt_raw @ W_dt + b_dt)
  dt_softplus_kernel<<<elemBlocks, 256, 0, stream>>>(dbl, W_dt, b_dt, dt);

  // 6. selective scan (1536 waves -> 192 blocks of 8 waves)
  scan_kernel<<<DINNER / 8, 256, 0, stream>>>(dt, dbl, xc, A_log, yb);

  // 7. gate + skip
  gate_kernel<<<elemBlocks, 256, 0, stream>>>(xz, xc, D_skip, yb);

  // 8. out = y @ W_out  (1024x1536 @ 1536x768): (1024/128)*(768/64) = 96 blocks
  wmma_gemm_blk<DINNER, DMODEL, DMODEL, DINNER>
      <<<8 * 12, 256, 0, stream>>>(yb, W_out, out, 12);
}